// Self_Attn_58609123721845
// MI455X (gfx1250) — compile-verified
//
#include <hip/hip_runtime.h>
#include <hip/hip_bf16.h>

typedef float v2f __attribute__((ext_vector_type(2)));
typedef float v8f __attribute__((ext_vector_type(8)));

#define B_   4
#define C_   128
#define W_   64
#define N_   4096
#define CQ_  16
#define NT_  256   // N_/16

// ---------------------------------------------------------------------------
// K1: QKV projection.  q[b][n][d] (row-major for A-frag reads),
// k[b][d][n], v[b][c][n].  x tile staged in LDS, coalesced.
// ---------------------------------------------------------------------------
__global__ __launch_bounds__(256) void qkv_kernel(
    const float* __restrict__ x,
    const float* __restrict__ wq, const float* __restrict__ bq,
    const float* __restrict__ wk, const float* __restrict__ bk,
    const float* __restrict__ wv, const float* __restrict__ bv,
    float* __restrict__ q, float* __restrict__ k, float* __restrict__ v)
{
    __shared__ float xs[C_ * 128];                    // [c][n] 64 KB
    const int tid = threadIdx.x;
    const int b   = blockIdx.x >> 5;                  // 32 n-tiles of 128 per batch
    const int n0  = (blockIdx.x & 31) * 128;
    const float* xb = x + (size_t)b * C_ * N_;

    for (int e = tid; e < C_ * 128; e += 256) {
        int c = e >> 7, nn = e & 127;
        xs[c * 128 + nn] = xb[(size_t)c * N_ + n0 + nn];
    }
    __syncthreads();

    const int n    = tid & 127;
    const int half = tid >> 7;
    // 160 output rows total: [0,16)=q, [16,32)=k, [32,160)=v.
    for (int r0 = half * 80; r0 < half * 80 + 80; r0 += 8) {
        const float* wmat; const float* bias; int rb;
        if (r0 < 16)      { wmat = wq; bias = bq; rb = r0; }
        else if (r0 < 32) { wmat = wk; bias = bk; rb = r0 - 16; }
        else              { wmat = wv; bias = bv; rb = r0 - 32; }
        float acc[8];
        #pragma unroll
        for (int i = 0; i < 8; ++i) acc[i] = bias[rb + i];
        for (int c = 0; c < C_; ++c) {
            float xv = xs[c * 128 + n];
            #pragma unroll
            for (int i = 0; i < 8; ++i)
                acc[i] = fmaf(wmat[(rb + i) * C_ + c], xv, acc[i]);
        }
        #pragma unroll
        for (int i = 0; i < 8; ++i) {
            int r = r0 + i;
            if (r < 16)
                q[((size_t)b * N_ + n0 + n) * CQ_ + r] = acc[i];
            else if (r < 32)
                k[((size_t)b * CQ_ + (r - 16)) * N_ + n0 + n] = acc[i];
            else
                v[((size_t)b * C_ + (r - 32)) * N_ + n0 + n] = acc[i];
        }
    }
}

// ---------------------------------------------------------------------------
// K2: energy + block-causal mask + softmax -> attention (written to d_out).
// Flash-style two-pass per (batch, 16-row tile); f32 WMMA 16x16x4 (K=Cq=16).
// ---------------------------------------------------------------------------
__global__ __launch_bounds__(256) void attn_kernel(
    const float* __restrict__ q, const float* __restrict__ k,
    float* __restrict__ attn)
{
    const int b    = blockIdx.x >> 8;
    const int ti   = blockIdx.x & 255;
    const int tid  = threadIdx.x;
    const int wid  = tid >> 5;          // wave 0..7 (wave32)
    const int lane = tid & 31;
    const int col  = lane & 15;         // A: M index, B: N index, C/D: col
    const int kh   = lane >> 4;         // lane half -> K pair select
    const int nTiles = (ti >> 2) * 4 + 4;   // allowed 16-col tiles
    const int Lim    = nTiles * 16;

    __shared__ float redM[8][16];
    __shared__ float redL[8][16];
    __shared__ float rowM[16];
    __shared__ float rowIL[16];

    // A fragments (Q tile), reused across all column tiles.
    const float* qrow = q + ((size_t)b * N_ + (size_t)ti * 16 + col) * CQ_;
    v2f aF[4];
    #pragma unroll
    for (int kc = 0; kc < 4; ++kc) {
        int d = kc * 4 + kh * 2;
        aF[kc].x = qrow[d];
        aF[kc].y = qrow[d + 1];
    }
    const float* kb = k + (size_t)b * CQ_ * N_;

    float m_[8], l_[8];
    #pragma unroll
    for (int i = 0; i < 8; ++i) { m_[i] = -__builtin_inff(); l_[i] = 0.0f; }

    // ---- pass 1: online row stats -------------------------------------
    for (int tj = wid; tj < nTiles; tj += 8) {
        v8f acc = {};
        #pragma unroll
        for (int kc = 0; kc < 4; ++kc) {
            int d = kc * 4 + kh * 2;
            v2f bF;
            bF.x = kb[(size_t)d * N_ + (size_t)tj * 16 + col];
            bF.y = kb[(size_t)(d + 1) * N_ + (size_t)tj * 16 + col];
            acc = __builtin_amdgcn_wmma_f32_16x16x4_f32(
                false, aF[kc], false, bF, (short)0, acc, false, false);
        }
        #pragma unroll
        for (int i = 0; i < 8; ++i) {      // row = i + 8*kh, 16 lanes/row
            float e = acc[i];
            float rmx = e;
            rmx = fmaxf(rmx, __shfl_xor(rmx, 1));
            rmx = fmaxf(rmx, __shfl_xor(rmx, 2));
            rmx = fmaxf(rmx, __shfl_xor(rmx, 4));
            rmx = fmaxf(rmx, __shfl_xor(rmx, 8));
            float mNew = fmaxf(m_[i], rmx);
            float p = __expf(e - mNew);
            p += __shfl_xor(p, 1);
            p += __shfl_xor(p, 2);
            p += __shfl_xor(p, 4);
            p += __shfl_xor(p, 8);
            l_[i] = l_[i] * __expf(m_[i] - mNew) + p;
            m_[i] = mNew;
        }
    }
    if (col == 0) {
        #pragma unroll
        for (int i = 0; i < 8; ++i) {
            redM[wid][i + 8 * kh] = m_[i];
            redL[wid][i + 8 * kh] = l_[i];
        }
    }
    __syncthreads();
    if (tid < 16) {
        float M = -__builtin_inff();
        #pragma unroll
        for (int w = 0; w < 8; ++w) M = fmaxf(M, redM[w][tid]);
        float L = 0.0f;
        #pragma unroll
        for (int w = 0; w < 8; ++w) L += redL[w][tid] * __expf(redM[w][tid] - M);
        rowM[tid]  = M;
        rowIL[tid] = 1.0f / L;             // L >= 1 (max term contributes 1)
    }
    __syncthreads();
    float Mv[8], Iv[8];
    #pragma unroll
    for (int i = 0; i < 8; ++i) { Mv[i] = rowM[i + 8 * kh]; Iv[i] = rowIL[i + 8 * kh]; }

    // ---- pass 2: recompute, normalize, store --------------------------
    float* arow = attn + ((size_t)b * N_ + (size_t)ti * 16) * (size_t)N_;
    for (int tj = wid; tj < nTiles; tj += 8) {
        v8f acc = {};
        #pragma unroll
        for (int kc = 0; kc < 4; ++kc) {
            int d = kc * 4 + kh * 2;
            v2f bF;
            bF.x = kb[(size_t)d * N_ + (size_t)tj * 16 + col];
            bF.y = kb[(size_t)(d + 1) * N_ + (size_t)tj * 16 + col];
            acc = __builtin_amdgcn_wmma_f32_16x16x4_f32(
                false, aF[kc], false, bF, (short)0, acc, false, false);
        }
        #pragma unroll
        for (int i = 0; i < 8; ++i) {
            float valv = __expf(acc[i] - Mv[i]) * Iv[i];
            arow[(size_t)(i + 8 * kh) * N_ + (size_t)tj * 16 + col] = valv;
        }
    }
    // ---- zero-fill masked region (softmax of -inf is exactly 0) -------
    const int Z = N_ - Lim;                // multiple of 64
    if (Z > 0) {
        const int z4 = Z >> 2;
        for (int e = tid; e < 16 * z4; e += 256) {
            int r = e / z4, c4 = e - r * z4;
            *(float4*)&arow[(size_t)r * N_ + Lim + c4 * 4] =
                make_float4(0.f, 0.f, 0.f, 0.f);
        }
    }
}

// ---------------------------------------------------------------------------
// K3: out = gamma * (V x attn^T) + x.  One block = 4 waves, one 16-col
// n-tile.  Tiles staged via GLOBAL_LOAD_ASYNC_TO_LDS_B128 (ASYNCcnt,
// VGPR-bypass) -- the CDNA5-native path for global->LDS staging.
// Row pad = 20 floats (80 B): keeps 16 B chunk alignment for B128 and
// conflict-free strided column reads (20*col mod 64 spans 16 banks).
// ---------------------------------------------------------------------------
typedef int v4i_ __attribute__((vector_size(4 * sizeof(int))));
typedef __attribute__((address_space(1))) v4i_ gv4;   // global int4
typedef __attribute__((address_space(3))) v4i_ lv4;   // LDS int4

__device__ __forceinline__ void async_copy16(const float* gsrc, float* ldst)
{
    __builtin_amdgcn_global_load_async_to_lds_b128(
        (gv4*)(__UINTPTR_TYPE__)gsrc,
        (lv4*)(__UINTPTR_TYPE__)ldst,
        0, 0);
}

__global__ __launch_bounds__(128) void av_kernel(
    const float* __restrict__ v, const float* __restrict__ attn,
    const float* __restrict__ x, const float* __restrict__ gamma,
    float* __restrict__ out)
{
    const int tn   = blockIdx.x & 255;
    const int cg   = (blockIdx.x >> 8) & 1;     // c-group: rows [cg*64, cg*64+64)
    const int b    = blockIdx.x >> 9;
    const int tid  = threadIdx.x;
    const int wid  = tid >> 5;                  // 0..3
    const int lane = tid & 31;
    const int col  = lane & 15;
    const int kh   = lane >> 4;

    __shared__ float at[16][20];                // attn tile  [n'][m']
    __shared__ float vs[4][16][20];             // V tiles    [wave][c'][m']

    const int mTiles = (tn >> 2) * 4 + 4;       // allowed K tiles for this n-tile
    const float* ab = attn + ((size_t)b * N_ + (size_t)tn * 16) * (size_t)N_;
    const float* vb = v + ((size_t)b * C_ + (size_t)cg * 64) * (size_t)N_;

    v8f acc = {};
    for (int mt = 0; mt < mTiles; ++mt) {
        const int m0 = mt * 16;
        if (mt + 1 < mTiles)
            __builtin_prefetch(ab + m0 + 16, 0, 1);   // global_prefetch_b8
        // attn tile: 64 x 16B chunks, one per thread (tid < 64)
        if (tid < 64) {
            int rr = tid >> 2, c4 = tid & 3;
            async_copy16(ab + (size_t)rr * N_ + m0 + c4 * 4, &at[rr][c4 * 4]);
        }
        // 4 V tiles: 256 x 16B chunks, two per thread
        #pragma unroll
        for (int it = 0; it < 2; ++it) {
            int e   = tid + it * 128;
            int wv2 = e >> 6, rr = (e & 63) >> 2, c4 = e & 3;
            async_copy16(vb + (size_t)(wv2 * 16 + rr) * N_ + m0 + c4 * 4,
                         &vs[wv2][rr][c4 * 4]);
        }
        __builtin_amdgcn_s_wait_asynccnt(0);
        __syncthreads();
        #pragma unroll
        for (int kc = 0; kc < 4; ++kc) {
            int d = kc * 4 + kh * 2;
            v2f aF, bF;
            aF.x = vs[wid][col][d];      // A: M=c'=lane&15, K=d
            aF.y = vs[wid][col][d + 1];
            bF.x = at[col][d];           // B[k][n'] = attn[n'][m0+k]
            bF.y = at[col][d + 1];
            acc = __builtin_amdgcn_wmma_f32_16x16x4_f32(
                false, aF, false, bF, (short)0, acc, false, false);
        }
        __syncthreads();
    }
    const float g = gamma[0];
    const size_t obase =
        ((size_t)b * C_ + (size_t)cg * 64 + (size_t)wid * 16) * (size_t)N_ +
        (size_t)tn * 16;
    #pragma unroll
    for (int i = 0; i < 8; ++i) {        // D row = i + 8*kh, col = lane&15
        size_t idx = obase + (size_t)(i + 8 * kh) * N_ + col;
        out[idx] = fmaf(g, acc[i], x[idx]);
    }
}

// ---------------------------------------------------------------------------
extern "C" void kernel_launch(void* const* d_in, const int* in_sizes, int n_in,
                              void* d_out, int out_size, void* d_ws, size_t ws_size,
                              hipStream_t stream)
{
    (void)in_sizes; (void)n_in; (void)out_size; (void)ws_size;
    const float* x     = (const float*)d_in[0];
    const float* wq    = (const float*)d_in[1];
    const float* bq    = (const float*)d_in[2];
    const float* wk    = (const float*)d_in[3];
    const float* bk    = (const float*)d_in[4];
    const float* wv    = (const float*)d_in[5];
    const float* bv    = (const float*)d_in[6];
    const float* gamma = (const float*)d_in[7];

    float* outp = (float*)d_out;                       // [B,C,W,H]
    float* attn = outp + (size_t)B_ * C_ * N_;         // [B,N,N]

    float* qbuf = (float*)d_ws;                        // [B,N,Cq]   1 MB
    float* kbuf = qbuf + (size_t)B_ * N_ * CQ_;        // [B,Cq,N]   1 MB
    float* vbuf = kbuf + (size_t)B_ * CQ_ * N_;        // [B,C,N]    8 MB

    qkv_kernel<<<B_ * (N_ / 128), 256, 0, stream>>>(
        x, wq, bq, wk, bk, wv, bv, qbuf, kbuf, vbuf);
    attn_kernel<<<B_ * NT_, 256, 0, stream>>>(qbuf, kbuf, attn);
    av_kernel<<<B_ * 2 * NT_, 128, 0, stream>>>(vbuf, attn, x, gamma, outp);
}